// GATEncoder_15410342658533
// MI455X (gfx1250) — compile-verified
//
#include <hip/hip_runtime.h>
#include <math.h>

typedef float v2f __attribute__((ext_vector_type(2)));
typedef float v8f __attribute__((ext_vector_type(8)));

// ---------- helpers ----------
__device__ __forceinline__ void atomicMaxFloat(float* addr, float val) {
    // IEEE-754 ordered-bits trick: works for mixed signs when initialized to -inf
    if (val >= 0.0f) {
        atomicMax((int*)addr, __float_as_int(val));
    } else {
        atomicMin((unsigned int*)addr, (unsigned int)__float_as_int(val));
    }
}

// ---------- fp32 WMMA GEMM: C[M,128] = A[M,128] @ B[128,128] ----------
// One wave32 computes one 16x16 tile via V_WMMA_F32_16X16X4_F32, K-loop of 32.
__global__ __launch_bounds__(32)
void gemm16_wmma_f32(const float* __restrict__ A, const float* __restrict__ B,
                     float* __restrict__ C, int M) {
    const int row0 = blockIdx.x << 4;
    const int col0 = blockIdx.y << 4;
    const int lane = threadIdx.x & 31;
    const int half = lane >> 4;   // 0 -> K pair {0,1}; 1 -> K pair {2,3}
    const int lo   = lane & 15;

    int arow_idx = row0 + lo;
    if (arow_idx >= M) arow_idx = M - 1;            // clamped read; store is guarded
    const float* arow = A + (size_t)arow_idx * 128;

    v8f acc = {0.f, 0.f, 0.f, 0.f, 0.f, 0.f, 0.f, 0.f};
#pragma unroll 8
    for (int k = 0; k < 128; k += 4) {
        v2f a;
        a[0] = arow[k + 2 * half + 0];
        a[1] = arow[k + 2 * half + 1];
        v2f b;
        b[0] = B[(size_t)(k + 2 * half + 0) * 128 + col0 + lo];
        b[1] = B[(size_t)(k + 2 * half + 1) * 128 + col0 + lo];
        // 8 args: (neg_a, A, neg_b, B, c_mod, C, reuse_a, reuse_b)
        acc = __builtin_amdgcn_wmma_f32_16x16x4_f32(false, a, false, b,
                                                    (short)0, acc, false, false);
    }
#pragma unroll
    for (int r = 0; r < 8; ++r) {
        int row = row0 + r + 8 * half;   // VGPR r: lanes0-15 -> M=r, lanes16-31 -> M=r+8
        if (row < M) C[(size_t)row * 128 + col0 + lo] = acc[r];
    }
}

// ---------- per-node attention scalars ----------
__global__ void node_att(const float* __restrict__ h,
                         const float* __restrict__ att_src,
                         const float* __restrict__ att_dst,
                         float* __restrict__ asrc, float* __restrict__ adst,
                         int n, int H, int out_c) {
    int t = blockIdx.x * blockDim.x + threadIdx.x;
    if (t >= n * H) return;
    int node = t / H, hh = t % H;
    const float* hp = h + (size_t)node * 128 + hh * out_c;
    float ss = 0.f, dd = 0.f;
    for (int c = 0; c < out_c; ++c) {
        float v = hp[c];
        ss += v * att_src[hh * out_c + c];
        dd += v * att_dst[hh * out_c + c];
    }
    asrc[t] = ss;
    adst[t] = dd;
}

// ---------- fill ----------
__global__ void fill_f32(float* __restrict__ p, float v, long long n) {
    long long t = (long long)blockIdx.x * blockDim.x + threadIdx.x;
    if (t < n) p[t] = v;
}

// ---------- edge pass 1: segment max of leaky_relu(a_src[s]+a_dst[d]) ----------
__global__ void edge_max(const int* __restrict__ ei, int E, int n, int H,
                         const float* __restrict__ asrc, const float* __restrict__ adst,
                         float* __restrict__ m) {
    long long t = (long long)blockIdx.x * blockDim.x + threadIdx.x;
    long long total = (long long)(E + n) * H;
    if (t >= total) return;
    int e = (int)(t / H), hh = (int)(t % H);
    int s, d;
    if (e < E) { s = ei[e]; d = ei[E + e]; } else { s = d = e - E; }
    float v = asrc[(size_t)s * H + hh] + adst[(size_t)d * H + hh];
    v = (v > 0.f) ? v : 0.2f * v;
    atomicMaxFloat(&m[(size_t)d * H + hh], v);
}

// ---------- edge pass 2: p = exp(e - m[dst]); den[dst] += p ----------
__global__ void edge_exp(const int* __restrict__ ei, int E, int n, int H,
                         const float* __restrict__ asrc, const float* __restrict__ adst,
                         const float* __restrict__ m, float* __restrict__ pe,
                         float* __restrict__ den) {
    long long t = (long long)blockIdx.x * blockDim.x + threadIdx.x;
    long long total = (long long)(E + n) * H;
    if (t >= total) return;
    int e = (int)(t / H), hh = (int)(t % H);
    int s, d;
    if (e < E) { s = ei[e]; d = ei[E + e]; } else { s = d = e - E; }
    float v = asrc[(size_t)s * H + hh] + adst[(size_t)d * H + hh];
    v = (v > 0.f) ? v : 0.2f * v;
    float p = expf(v - m[(size_t)d * H + hh]);
    pe[t] = p;
    atomicAdd(&den[(size_t)d * H + hh], p);
}

// ---------- edge pass 3: out[dst] += alpha * h[src], 4 channels per thread ----------
__global__ void edge_scatter(const int* __restrict__ ei, int E, int n, int H, int out_c,
                             const float* __restrict__ pe, const float* __restrict__ den,
                             const float* __restrict__ h, float* __restrict__ out) {
    long long t = (long long)blockIdx.x * blockDim.x + threadIdx.x;
    long long total = (long long)(E + n) * 32;   // 128 channels / 4 per thread
    if (t >= total) return;
    int e = (int)(t >> 5);
    int c = (int)(t & 31) * 4;
    int s, d;
    if (e < E) { s = ei[e]; d = ei[E + e]; } else { s = d = e - E; }
    int hh = c / out_c;
    float alpha = pe[(size_t)e * H + hh] / (den[(size_t)d * H + hh] + 1e-16f);
    const float4 hv = *(const float4*)(h + (size_t)s * 128 + c);
    float* o = out + (size_t)d * 128 + c;
    atomicAdd(o + 0, hv.x * alpha);
    atomicAdd(o + 1, hv.y * alpha);
    atomicAdd(o + 2, hv.z * alpha);
    atomicAdd(o + 3, hv.w * alpha);
}

// ---------- bias (+ optional ELU) ----------
__global__ void bias_act(float* __restrict__ out, const float* __restrict__ bias,
                         long long total, int do_elu) {
    long long t = (long long)blockIdx.x * blockDim.x + threadIdx.x;
    if (t >= total) return;
    float v = out[t] + bias[(int)(t & 127)];
    if (do_elu) v = (v > 0.f) ? v : expm1f(v);
    out[t] = v;
}

// ---------- driver ----------
static inline unsigned int cdiv(long long a, int b) { return (unsigned int)((a + b - 1) / b); }

extern "C" void kernel_launch(void* const* d_in, const int* in_sizes, int n_in,
                              void* d_out, int out_size, void* d_ws, size_t ws_size,
                              hipStream_t stream) {
    const float* x   = (const float*)d_in[0];
    const int*   ei  = (const int*)d_in[1];
    const float* W1  = (const float*)d_in[2];
    const float* as1 = (const float*)d_in[3];
    const float* ad1 = (const float*)d_in[4];
    const float* b1  = (const float*)d_in[5];
    const float* W2  = (const float*)d_in[6];
    const float* as2 = (const float*)d_in[7];
    const float* ad2 = (const float*)d_in[8];
    const float* b2  = (const float*)d_in[9];
    float* out = (float*)d_out;

    const int N = in_sizes[0] / 128;
    const int E = in_sizes[1] / 2;
    const long long NE = (long long)N + E;

    // workspace layout (floats)
    float* ws   = (float*)d_ws;
    float* hbuf = ws;                              // N*128 (reused both layers)
    float* x2   = hbuf + (size_t)N * 128;          // N*128 (layer-1 output / layer-2 input)
    float* asrc = x2   + (size_t)N * 128;          // N*4
    float* adst = asrc + (size_t)N * 4;            // N*4
    float* mbuf = adst + (size_t)N * 4;            // N*4
    float* den  = mbuf + (size_t)N * 4;            // N*4
    float* pe   = den  + (size_t)N * 4;            // (E+N)*4

    const dim3 gemmGrid(cdiv(N, 16), 8);
    const int TB = 256;
    const float NEG_INF = -INFINITY;

    // ================= Layer 1 (H=4, out_c=32) =================
    gemm16_wmma_f32<<<gemmGrid, 32, 0, stream>>>(x, W1, hbuf, N);
    node_att<<<cdiv((long long)N * 4, TB), TB, 0, stream>>>(hbuf, as1, ad1, asrc, adst, N, 4, 32);
    fill_f32<<<cdiv((long long)N * 4, TB), TB, 0, stream>>>(mbuf, NEG_INF, (long long)N * 4);
    fill_f32<<<cdiv((long long)N * 4, TB), TB, 0, stream>>>(den, 0.f, (long long)N * 4);
    fill_f32<<<cdiv((long long)N * 128, TB), TB, 0, stream>>>(x2, 0.f, (long long)N * 128);
    edge_max<<<cdiv(NE * 4, TB), TB, 0, stream>>>(ei, E, N, 4, asrc, adst, mbuf);
    edge_exp<<<cdiv(NE * 4, TB), TB, 0, stream>>>(ei, E, N, 4, asrc, adst, mbuf, pe, den);
    edge_scatter<<<cdiv(NE * 32, TB), TB, 0, stream>>>(ei, E, N, 4, 32, pe, den, hbuf, x2);
    bias_act<<<cdiv((long long)N * 128, TB), TB, 0, stream>>>(x2, b1, (long long)N * 128, 1);

    // ================= Layer 2 (H=1, out_c=128) =================
    gemm16_wmma_f32<<<gemmGrid, 32, 0, stream>>>(x2, W2, hbuf, N);
    node_att<<<cdiv((long long)N, TB), TB, 0, stream>>>(hbuf, as2, ad2, asrc, adst, N, 1, 128);
    fill_f32<<<cdiv((long long)N, TB), TB, 0, stream>>>(mbuf, NEG_INF, (long long)N);
    fill_f32<<<cdiv((long long)N, TB), TB, 0, stream>>>(den, 0.f, (long long)N);
    fill_f32<<<cdiv((long long)N * 128, TB), TB, 0, stream>>>(out, 0.f, (long long)N * 128);
    edge_max<<<cdiv(NE, TB), TB, 0, stream>>>(ei, E, N, 1, asrc, adst, mbuf);
    edge_exp<<<cdiv(NE, TB), TB, 0, stream>>>(ei, E, N, 1, asrc, adst, mbuf, pe, den);
    edge_scatter<<<cdiv(NE * 32, TB), TB, 0, stream>>>(ei, E, N, 1, 128, pe, den, hbuf, out);
    bias_act<<<cdiv((long long)N * 128, TB), TB, 0, stream>>>(out, b2, (long long)N * 128, 0);
}